// PointTransformerLayer_77060303224836
// MI455X (gfx1250) — compile-verified
//
#include <hip/hip_runtime.h>
#include <hip/hip_bf16.h>

// Point Transformer layer for MI455X (gfx1250, wave32, WMMA).
// B=4, N=4096, DIM=128, K=16 (hardcoded from the reference).
//
// Pipeline:
//   0) repack kernels:  weights -> (K/2, N) float2 "K-pair" layout so each
//                       WMMA B-fragment is ONE global_load_b64 (L2-resident)
//   1) knn_kernel:      brute-force kNN (register insertion-sort top-16)
//   2) qproj_kernel:    Q = X @ qw + qb            (WMMA f32 16x16x4)
//   3) fused_attn:      one workgroup (8 waves) per query point:
//        gather 16 neighbors -> LDS
//        pe1 = relu(rel@pw1+pb1)    WMMA, K=4 (zero-padded), 1 wmma/wave
//        pe/kk/v GEMMs              interleaved triple WMMA chain, 96 wmma/wave-tile
//        a   = q - kk + pe
//        h   = relu(a@aw1+ab1)      WMMA on waves 0-1 (16x32, K=128)
//        att = h@aw2+ab2            WMMA, K=32
//        softmax over 16 neighbors per channel, out = sum att*(v+pe)
// All per-query intermediates live in LDS (~54 KB/WG of the 320 KB WGP pool).

#define B_    4
#define N_    4096
#define DIM_  128
#define K_    16
#define SSTR  132   // LDS row stride (dwords); 132 % 64 == 4 -> conflict-free frags

typedef __attribute__((ext_vector_type(2))) float v2f;
typedef __attribute__((ext_vector_type(8))) float v8f;

static __device__ __forceinline__ v8f wmma4(v2f a, v2f b, v8f c) {
  // D = A(16x4 f32) * B(4x16 f32) + C(16x16 f32), full f32 precision.
  return __builtin_amdgcn_wmma_f32_16x16x4_f32(
      /*neg_a=*/false, a, /*neg_b=*/false, b,
      /*c_mod=*/(short)0, c, /*reuse_a=*/false, /*reuse_b=*/false);
}

static __device__ __forceinline__ v2f ld2s(const float* p) {
  const float2 t = *(const float2*)p;
  v2f r; r.x = t.x; r.y = t.y; return r;
}
static __device__ __forceinline__ v2f ld2g(const float2* __restrict__ p) {
  const float2 t = *p;
  v2f r; r.x = t.x; r.y = t.y; return r;
}

// ---------------------------------------------------------------------------
// Weight repack: W (K x ncols, row-major f32) -> Wp[t][col] = {W[2t][col], W[2t+1][col]}
// so B-fragment for K-step kb, lane-half hk is the single float2 Wp[kb*2+hk][colN].
// ---------------------------------------------------------------------------
__global__ __launch_bounds__(256) void repack_pair_kernel(
    const float* __restrict__ W, float2* __restrict__ Wp, int halfK, int ncols) {
  const int gid = blockIdx.x * 256 + threadIdx.x;
  if (gid >= halfK * ncols) return;
  const int t = gid / ncols, col = gid % ncols;
  Wp[gid] = make_float2(W[(2 * t) * ncols + col], W[(2 * t + 1) * ncols + col]);
}

// pw1 is (3 x 128): zero-pad K to 4 -> 2 pair-rows.
__global__ __launch_bounds__(256) void repack_pw1_kernel(
    const float* __restrict__ W, float2* __restrict__ Wp) {
  const int gid = threadIdx.x;         // 0..255
  const int t = gid >> 7, col = gid & 127;
  const float lo = W[(2 * t) * DIM_ + col];
  const float hi = (2 * t + 1 < 3) ? W[(2 * t + 1) * DIM_ + col] : 0.f;
  Wp[gid] = make_float2(lo, hi);
}

// ---------------------------------------------------------------------------
// Kernel 1: brute-force kNN. One thread per query, candidate tiles in LDS.
// dist = |pi|^2 + |pj|^2 - 2 pi.pj  (same formula/order as the reference).
// ---------------------------------------------------------------------------
__global__ __launch_bounds__(128) void knn_kernel(const float* __restrict__ pos,
                                                  int* __restrict__ idx_out) {
  __shared__ float sP[128 * 3];
  __shared__ float sXX[128];
  const int t = threadIdx.x;
  const int bpb = N_ / 128;
  const int b = blockIdx.x / bpb;
  const int q = (blockIdx.x % bpb) * 128 + t;

  const float qx = pos[(b * N_ + q) * 3 + 0];
  const float qy = pos[(b * N_ + q) * 3 + 1];
  const float qz = pos[(b * N_ + q) * 3 + 2];
  const float qxx = qx * qx + qy * qy + qz * qz;

  float bd[K_];
  int   bi[K_];
#pragma unroll
  for (int i = 0; i < K_; ++i) { bd[i] = __builtin_inff(); bi[i] = 0; }

  for (int m0 = 0; m0 < N_; m0 += 128) {
    __syncthreads();
    const float px = pos[(b * N_ + m0 + t) * 3 + 0];
    const float py = pos[(b * N_ + m0 + t) * 3 + 1];
    const float pz = pos[(b * N_ + m0 + t) * 3 + 2];
    sP[t * 3 + 0] = px; sP[t * 3 + 1] = py; sP[t * 3 + 2] = pz;
    sXX[t] = px * px + py * py + pz * pz;
    __syncthreads();
    for (int mm = 0; mm < 128; ++mm) {
      const float dot = qx * sP[mm * 3 + 0] + qy * sP[mm * 3 + 1] + qz * sP[mm * 3 + 2];
      const float d = qxx + sXX[mm] - 2.0f * dot;
      if (d < bd[K_ - 1]) {                 // strict <: earlier index wins ties
        bd[K_ - 1] = d; bi[K_ - 1] = m0 + mm;
#pragma unroll
        for (int j = K_ - 1; j > 0; --j) {
          if (bd[j] < bd[j - 1]) {
            const float td = bd[j]; bd[j] = bd[j - 1]; bd[j - 1] = td;
            const int   ti = bi[j]; bi[j] = bi[j - 1]; bi[j - 1] = ti;
          }
        }
      }
    }
  }
#pragma unroll
  for (int i = 0; i < K_; ++i) idx_out[(b * N_ + q) * K_ + i] = bi[i];
}

// ---------------------------------------------------------------------------
// Kernel 2: Q = X @ qw + qb.  One block = 16 rows; 8 waves = 8 column tiles.
// ---------------------------------------------------------------------------
__global__ __launch_bounds__(256) void qproj_kernel(const float* __restrict__ x,
                                                    const float2* __restrict__ qwp,
                                                    const float* __restrict__ qb,
                                                    float* __restrict__ qout) {
  const int lane = threadIdx.x & 31;
  const int wv   = threadIdx.x >> 5;
  const int m    = lane & 15;
  const int hk   = lane >> 4;
  const int row0 = blockIdx.x * 16;
  const int colN = wv * 16 + m;

  v8f c = {0.f, 0.f, 0.f, 0.f, 0.f, 0.f, 0.f, 0.f};
#pragma unroll
  for (int kb = 0; kb < 32; ++kb) {
    const int k = kb * 4 + hk * 2;
    const v2f a = ld2s(&x[(row0 + m) * DIM_ + k]);
    const v2f b = ld2g(&qwp[(kb * 2 + hk) * DIM_ + colN]);
    c = wmma4(a, b, c);
  }
  const float bias = qb[colN];
#pragma unroll
  for (int i = 0; i < 8; ++i)
    qout[(row0 + hk * 8 + i) * DIM_ + colN] = c[i] + bias;
}

// ---------------------------------------------------------------------------
// Kernel 3: fused per-query attention. 256 threads = 8 waves per query.
// ---------------------------------------------------------------------------
__global__ __launch_bounds__(256) void fused_attn_kernel(
    const float* __restrict__ x,    const float* __restrict__ pos,
    const float2* __restrict__ pw1p, const float* __restrict__ pb1,
    const float2* __restrict__ pw2p, const float* __restrict__ pb2,
    const float2* __restrict__ kwp,  const float* __restrict__ kb_,
    const float2* __restrict__ vwp,  const float* __restrict__ vb,
    const float2* __restrict__ aw1p, const float* __restrict__ ab1,
    const float2* __restrict__ aw2p, const float* __restrict__ ab2,
    const int* __restrict__ knn,     const float* __restrict__ qbuf,
    float* __restrict__ out) {
  __shared__ float sNbr[16 * SSTR];   // gathered neighbor features
  __shared__ float sPe1[16 * SSTR];   // relu(rel@pw1+pb1)
  __shared__ float sPe [16 * SSTR];   // pe
  __shared__ float sAm [16 * SSTR];   // a = q - kk + pe
  __shared__ float sV  [16 * SSTR];   // v
  __shared__ float sAt [16 * SSTR];   // attention logits
  __shared__ float sH  [16 * 36];     // hidden (16x32, padded stride)
  __shared__ float sQ  [DIM_];
  __shared__ float sRel[16 * 4];      // rel pos, K padded 3->4 with zeros
  __shared__ int   sIdx[16];

  const int tid  = threadIdx.x;
  const int b    = blockIdx.x / N_;
  const int n    = blockIdx.x % N_;
  const int lane = tid & 31;
  const int wv   = tid >> 5;
  const int m    = lane & 15;
  const int hk   = lane >> 4;
  const int colN = wv * 16 + m;

  if (tid < 16) sIdx[tid] = knn[(b * N_ + n) * K_ + tid];
  if (tid >= 16 && tid < 20) {          // L2 warm-up for the streamed weights
    const float2* wp = (tid == 16) ? pw2p : (tid == 17) ? kwp : (tid == 18) ? vwp : aw2p;
    __builtin_prefetch(wp, 0, 0);
  }
  __syncthreads();

  // Gather 16x128 neighbor features (float4, 2 per thread) + q row + rel pos.
#pragma unroll
  for (int i = 0; i < 2; ++i) {
    const int v  = tid + 256 * i;       // 0..511
    const int r  = v >> 5;              // neighbor row
    const int c4 = v & 31;              // float4 column
    const float4 f = *(const float4*)&x[(b * N_ + sIdx[r]) * DIM_ + c4 * 4];
    *(float4*)&sNbr[r * SSTR + c4 * 4] = f;
  }
  if (tid < DIM_) sQ[tid] = qbuf[(b * N_ + n) * DIM_ + tid];
  if (tid >= 128 && tid < 192) {        // 64 threads: 16 rows x 4 (padded) comps
    const int t2 = tid - 128;
    const int r = t2 >> 2, j = t2 & 3;
    sRel[t2] = (j < 3)
        ? pos[(b * N_ + n) * 3 + j] - pos[(b * N_ + sIdx[r]) * 3 + j]
        : 0.f;
  }
  __syncthreads();

  // pe1 = relu(rel @ pw1 + pb1) : (16,128), K=3 zero-padded to 4 -> 1 WMMA/wave.
  {
    v2f a; a.x = sRel[m * 4 + hk * 2]; a.y = sRel[m * 4 + hk * 2 + 1];
    const v2f bb = ld2g(&pw1p[hk * DIM_ + colN]);
    v8f c = {0.f, 0.f, 0.f, 0.f, 0.f, 0.f, 0.f, 0.f};
    c = wmma4(a, bb, c);
    const float bias = pb1[colN];
#pragma unroll
    for (int i = 0; i < 8; ++i)
      sPe1[(hk * 8 + i) * SSTR + colN] = fmaxf(c[i] + bias, 0.f);
  }
  __syncthreads();

  // Three 16x128x128 GEMMs, interleaved -> 3 independent WMMA chains per wave.
  {
    v8f c_pe = {0.f, 0.f, 0.f, 0.f, 0.f, 0.f, 0.f, 0.f};
    v8f c_kk = c_pe, c_v = c_pe;
#pragma unroll
    for (int kb = 0; kb < 32; ++kb) {
      const int k = kb * 4 + hk * 2;
      const int t = kb * 2 + hk;
      const v2f a_pe = ld2s(&sPe1[m * SSTR + k]);
      const v2f a_nb = ld2s(&sNbr[m * SSTR + k]);
      const v2f b_pe = ld2g(&pw2p[t * DIM_ + colN]);
      const v2f b_kk = ld2g(&kwp [t * DIM_ + colN]);
      const v2f b_v  = ld2g(&vwp [t * DIM_ + colN]);
      c_pe = wmma4(a_pe, b_pe, c_pe);
      c_kk = wmma4(a_nb, b_kk, c_kk);
      c_v  = wmma4(a_nb, b_v , c_v );
    }
    const float bpe = pb2[colN], bkk = kb_[colN], bv = vb[colN], qv = sQ[colN];
#pragma unroll
    for (int i = 0; i < 8; ++i) {
      const int r = hk * 8 + i;
      const float pe  = c_pe[i] + bpe;
      const float kkv = c_kk[i] + bkk;
      sPe[r * SSTR + colN] = pe;
      sV [r * SSTR + colN] = c_v[i] + bv;
      sAm[r * SSTR + colN] = qv - kkv + pe;   // a = q - k + pe
    }
  }
  __syncthreads();

  // h = relu(a @ aw1 + ab1) : (16,32), K=128 -> 2 WMMA tiles on waves 0-1.
  if (wv < 2) {
    const int colH = wv * 16 + m;            // 0..31
    v8f c = {0.f, 0.f, 0.f, 0.f, 0.f, 0.f, 0.f, 0.f};
#pragma unroll
    for (int kb = 0; kb < 32; ++kb) {
      const int k = kb * 4 + hk * 2;
      const v2f a = ld2s(&sAm[m * SSTR + k]);
      const v2f bb = ld2g(&aw1p[(kb * 2 + hk) * 32 + colH]);
      c = wmma4(a, bb, c);
    }
    const float bias = ab1[colH];
#pragma unroll
    for (int i = 0; i < 8; ++i)
      sH[(hk * 8 + i) * 36 + colH] = fmaxf(c[i] + bias, 0.f);
  }
  __syncthreads();

  // attn = h @ aw2 + ab2 : (16,128), K=32 -> 8 WMMA K-steps per wave.
  {
    v8f c = {0.f, 0.f, 0.f, 0.f, 0.f, 0.f, 0.f, 0.f};
#pragma unroll
    for (int kb = 0; kb < 8; ++kb) {
      const int k = kb * 4 + hk * 2;
      v2f a; a.x = sH[m * 36 + k]; a.y = sH[m * 36 + k + 1];
      const v2f bb = ld2g(&aw2p[(kb * 2 + hk) * DIM_ + colN]);
      c = wmma4(a, bb, c);
    }
    const float bat = ab2[colN];
#pragma unroll
    for (int i = 0; i < 8; ++i) sAt[(hk * 8 + i) * SSTR + colN] = c[i] + bat;
  }
  __syncthreads();

  // Softmax over the 16 neighbors per channel; out = sum attn*(v+pe).
  if (tid < DIM_) {
    float mx = -__builtin_inff();
#pragma unroll
    for (int r = 0; r < 16; ++r) mx = fmaxf(mx, sAt[r * SSTR + tid]);
    float ssum = 0.f, acc = 0.f;
#pragma unroll
    for (int r = 0; r < 16; ++r) {
      const float e = __expf(sAt[r * SSTR + tid] - mx);
      ssum += e;
      acc  += e * (sV[r * SSTR + tid] + sPe[r * SSTR + tid]);
    }
    out[(b * N_ + n) * DIM_ + tid] = acc / ssum;
  }
}

// ---------------------------------------------------------------------------
extern "C" void kernel_launch(void* const* d_in, const int* in_sizes, int n_in,
                              void* d_out, int out_size, void* d_ws, size_t ws_size,
                              hipStream_t stream) {
  (void)in_sizes; (void)n_in; (void)out_size; (void)ws_size;
  const float* x   = (const float*)d_in[0];
  const float* pos = (const float*)d_in[1];
  const float* pw1 = (const float*)d_in[2];
  const float* pb1 = (const float*)d_in[3];
  const float* pw2 = (const float*)d_in[4];
  const float* pb2 = (const float*)d_in[5];
  const float* qw  = (const float*)d_in[6];
  const float* qb  = (const float*)d_in[7];
  const float* kw  = (const float*)d_in[8];
  const float* kb  = (const float*)d_in[9];
  const float* vw  = (const float*)d_in[10];
  const float* vb  = (const float*)d_in[11];
  const float* aw1 = (const float*)d_in[12];
  const float* ab1 = (const float*)d_in[13];
  const float* aw2 = (const float*)d_in[14];
  const float* ab2 = (const float*)d_in[15];
  float* out = (float*)d_out;

  // Workspace carve-out (256B aligned slices).
  char* wsp = (char*)d_ws;
  size_t off = 0;
  auto carve = [&](size_t bytes) {
    char* p = wsp + off;
    off = (off + bytes + 255) & ~(size_t)255;
    return p;
  };
  int*    ws_idx = (int*)   carve((size_t)B_ * N_ * K_ * sizeof(int));
  float*  ws_q   = (float*) carve((size_t)B_ * N_ * DIM_ * sizeof(float));
  float2* pw1p   = (float2*)carve(2 * DIM_ * sizeof(float2));
  float2* pw2p   = (float2*)carve(64 * DIM_ * sizeof(float2));
  float2* kwp    = (float2*)carve(64 * DIM_ * sizeof(float2));
  float2* vwp    = (float2*)carve(64 * DIM_ * sizeof(float2));
  float2* qwp    = (float2*)carve(64 * DIM_ * sizeof(float2));
  float2* aw1p   = (float2*)carve(64 * 32 * sizeof(float2));
  float2* aw2p   = (float2*)carve(16 * DIM_ * sizeof(float2));

  // Weight repack (tiny; runs before the GEMM consumers on the same stream).
  repack_pw1_kernel<<<1, 256, 0, stream>>>(pw1, pw1p);
  repack_pair_kernel<<<(64 * DIM_ + 255) / 256, 256, 0, stream>>>(pw2, pw2p, 64, DIM_);
  repack_pair_kernel<<<(64 * DIM_ + 255) / 256, 256, 0, stream>>>(kw,  kwp,  64, DIM_);
  repack_pair_kernel<<<(64 * DIM_ + 255) / 256, 256, 0, stream>>>(vw,  vwp,  64, DIM_);
  repack_pair_kernel<<<(64 * DIM_ + 255) / 256, 256, 0, stream>>>(qw,  qwp,  64, DIM_);
  repack_pair_kernel<<<(64 * 32   + 255) / 256, 256, 0, stream>>>(aw1, aw1p, 64, 32);
  repack_pair_kernel<<<(16 * DIM_ + 255) / 256, 256, 0, stream>>>(aw2, aw2p, 16, DIM_);

  knn_kernel<<<(B_ * N_) / 128, 128, 0, stream>>>(pos, ws_idx);
  qproj_kernel<<<(B_ * N_) / 16, 256, 0, stream>>>(x, qwp, qb, ws_q);
  fused_attn_kernel<<<B_ * N_, 256, 0, stream>>>(
      x, pos, pw1p, pb1, pw2p, pb2, kwp, kb, vwp, vb, aw1p, ab1, aw2p, ab2,
      ws_idx, ws_q, out);
}